// GCN_57990648431249
// MI455X (gfx1250) — compile-verified
//
#include <hip/hip_runtime.h>
#include <hip/hip_bf16.h>

typedef float v2f __attribute__((ext_vector_type(2)));
typedef float v8f __attribute__((ext_vector_type(8)));
typedef int   v4i __attribute__((vector_size(16)));

#define N_NODES 50000
#define N_EDGES 800000
#define DIM     128

// gfx1250 async global->LDS copy path (ASYNCcnt-tracked), if this toolchain
// declares the builtins; otherwise fall back to load/ds_store staging.
#if defined(__has_builtin)
#if __has_builtin(__builtin_amdgcn_global_load_async_to_lds_b128) && \
    __has_builtin(__builtin_amdgcn_s_wait_asynccnt)
#define USE_ASYNC_LDS 1
#endif
#endif
#ifndef USE_ASYNC_LDS
#define USE_ASYNC_LDS 0
#endif

#if USE_ASYNC_LDS
typedef __attribute__((address_space(1))) v4i* gv4i_p;  // prints as "__device__"
typedef __attribute__((address_space(3))) v4i* lv4i_p;  // prints as "__shared__"

__device__ __forceinline__ void async_cp16(const float* g, float* l) {
    // b128 global -> LDS DMA, tracked by ASYNCcnt (no VGPR round trip)
    __builtin_amdgcn_global_load_async_to_lds_b128(
        (gv4i_p)(v4i*)g, (lv4i_p)(v4i*)l, 0, 0);
}
#endif

// ---------------------------------------------------------------------------
// Degree / normalization kernels
// ---------------------------------------------------------------------------
__global__ void k_deg_init(float* __restrict__ deg, int n) {
    int i = blockIdx.x * blockDim.x + threadIdx.x;
    if (i < n) deg[i] = 2.0f;   // improved=True self-loop weight
}

__global__ void k_deg_edges(const long long* __restrict__ ei,
                            const float* __restrict__ ew,
                            float* __restrict__ deg, int E) {
    int e = blockIdx.x * blockDim.x + threadIdx.x;
    if (e < E) {
        long long c = ei[(long long)E + e];   // col
        unsafeAtomicAdd(&deg[c], ew[e]);
    }
}

__global__ void k_dinv(float* __restrict__ deg /*in: deg, out: dinv*/,
                       float* __restrict__ selfnorm, int n) {
    int i = blockIdx.x * blockDim.x + threadIdx.x;
    if (i < n) {
        float d = deg[i];
        float r = (d > 0.0f) ? rsqrtf(d) : 0.0f;
        deg[i] = r;
        selfnorm[i] = 2.0f * r * r;          // dinv * 2.0 * dinv for self loop
    }
}

__global__ void k_norm(const long long* __restrict__ ei,
                       const float* __restrict__ ew,
                       const float* __restrict__ dinv,
                       float* __restrict__ nrm, int E) {
    int e = blockIdx.x * blockDim.x + threadIdx.x;
    if (e < E) {
        long long r = ei[e];
        long long c = ei[(long long)E + e];
        nrm[e] = dinv[r] * ew[e] * dinv[c];
    }
}

__global__ void k_zero(float* __restrict__ p, int n) {
    int i = blockIdx.x * blockDim.x + threadIdx.x;
    if (i < n) p[i] = 0.0f;
}

// ---------------------------------------------------------------------------
// Dense GEMM  Y[16rows x 128] = X_tile @ W  via V_WMMA_F32_16X16X4_F32.
// Block = 256 threads = 8 wave32s; wave w computes the 16x16 tile at col 16*w.
// Tiles staged into LDS with GLOBAL_LOAD_ASYNC_TO_LDS_B128 (ASYNCcnt) when
// available: no VGPR round trip for the staging traffic.
// ---------------------------------------------------------------------------
__global__ __launch_bounds__(256) void gemm16_wmma(const float* __restrict__ X,
                                                   const float* __restrict__ W,
                                                   float* __restrict__ Y) {
    __shared__ float lA[16][132];   // 132 floats = 528B row stride: 16B-aligned
                                    // rows AND conflict-free A-frag column reads
    __shared__ float lW[32][128];   // K-chunk of W

    const int t    = threadIdx.x;
    const int lane = t & 31;
    const int wave = t >> 5;
    const int row0 = blockIdx.x << 4;

    // ---- Stage the 16x128 A tile once (3125 blocks * 16 rows == 50000). ----
#if USE_ASYNC_LDS
    // 16 rows * 32 b128-chunks = 512 chunks; 2 per thread.
#pragma unroll
    for (int j = 0; j < 2; ++j) {
        int ci = t + (j << 8);
        int r = ci >> 5, c4 = (ci & 31) << 2;
        async_cp16(X + (size_t)(row0 + r) * DIM + c4, &lA[r][c4]);
    }
#else
    for (int i = t; i < 16 * DIM; i += 256) {
        lA[i >> 7][i & 127] = X[(size_t)(row0 + (i >> 7)) * DIM + (i & 127)];
    }
#endif

    const int m    = lane & 15;                 // A row within tile
    const int n    = (wave << 4) | (lane & 15); // output column 0..127
    const int koff = (lane >> 4) << 1;          // lanes 16-31 hold K+2,K+3

    v8f acc = {};
    for (int kb = 0; kb < DIM; kb += 32) {
        __syncthreads();                        // lW reuse hazard / lA ready
        // ---- Stage W[kb..kb+31][0..127]: 1024 b128-chunks, 4 per thread ----
#if USE_ASYNC_LDS
#pragma unroll
        for (int j = 0; j < 4; ++j) {
            int ci = t + (j << 8);
            int r = ci >> 5, c4 = (ci & 31) << 2;
            async_cp16(W + (size_t)(kb + r) * DIM + c4, &lW[r][c4]);
        }
        __builtin_amdgcn_s_wait_asynccnt(0);    // my wave's copies landed
#else
        for (int i = t; i < 32 * DIM; i += 256) {
            lW[i >> 7][i & 127] = W[(size_t)(kb + (i >> 7)) * DIM + (i & 127)];
        }
#endif
        __syncthreads();                        // everyone's copies landed
#pragma unroll
        for (int k = 0; k < 32; k += 4) {
            v2f a, b;
            a.x = lA[m][kb + k + koff];
            a.y = lA[m][kb + k + koff + 1];
            b.x = lW[k + koff][n];
            b.y = lW[k + koff + 1][n];
            acc = __builtin_amdgcn_wmma_f32_16x16x4_f32(
                false, a, false, b, (short)0, acc, false, false);
        }
    }

    // C/D layout: VGPR j -> lanes 0-15: M=j, N=lane; lanes 16-31: M=j+8.
    const int mrow = row0 + ((lane >> 4) << 3);
#pragma unroll
    for (int j = 0; j < 8; ++j) {
        Y[(size_t)(mrow + j) * DIM + n] = acc[j];
    }
}

// ---------------------------------------------------------------------------
// Edge aggregation: one wave per edge, float4 per lane (512B coalesced gather,
// hardware f32 atomic scatter into the L2 atomic units).
// ---------------------------------------------------------------------------
__global__ __launch_bounds__(256) void k_aggregate(const float* __restrict__ h,
                                                   const long long* __restrict__ ei,
                                                   const float* __restrict__ nrm,
                                                   float* __restrict__ acc, int E) {
    int gtid = blockIdx.x * blockDim.x + threadIdx.x;
    int e    = gtid >> 5;
    int lane = gtid & 31;
    if (e >= E) return;
    long long r = ei[e];
    long long c = ei[(long long)E + e];
    float w = nrm[e];
    const float4 hv = *(const float4*)(h + (size_t)r * DIM + lane * 4);
    float* dst = acc + (size_t)c * DIM + lane * 4;
    unsafeAtomicAdd(dst + 0, w * hv.x);
    unsafeAtomicAdd(dst + 1, w * hv.y);
    unsafeAtomicAdd(dst + 2, w * hv.z);
    unsafeAtomicAdd(dst + 3, w * hv.w);
}

// ---------------------------------------------------------------------------
// Epilogues
// ---------------------------------------------------------------------------
// Layer 1: h <- relu(acc + selfnorm*h + b)   (h holds x@W1 in, layer-out out)
__global__ void k_bias_relu(float* __restrict__ h,
                            const float* __restrict__ acc,
                            const float* __restrict__ selfnorm,
                            const float* __restrict__ b, int total) {
    int idx = blockIdx.x * blockDim.x + threadIdx.x;
    if (idx < total) {
        int node = idx >> 7, d = idx & 127;
        float v = acc[idx] + selfnorm[node] * h[idx] + b[d];
        h[idx] = v > 0.0f ? v : 0.0f;
    }
}

__device__ inline unsigned hash_u32(unsigned x) {
    x ^= x >> 16; x *= 0x7feb352du;
    x ^= x >> 15; x *= 0x846ca68bu;
    x ^= x >> 16; return x;
}

// Layer 2 + dropout: out <- dropout(relu(agg + selfnorm*g + b2))
__global__ void k_final(float* __restrict__ out,
                        const float* __restrict__ agg,   // scatter result
                        const float* __restrict__ g,     // h1@W2
                        const float* __restrict__ selfnorm,
                        const float* __restrict__ b, int total) {
    int idx = blockIdx.x * blockDim.x + threadIdx.x;
    if (idx < total) {
        int node = idx >> 7, d = idx & 127;
        float v = agg[idx] + selfnorm[node] * g[idx] + b[d];
        v = v > 0.0f ? v : 0.0f;
        // deterministic p=0.5 dropout, scale by 1/(1-p)=2
        v = (hash_u32((unsigned)idx) & 1u) ? (2.0f * v) : 0.0f;
        out[idx] = v;
    }
}

// ---------------------------------------------------------------------------
extern "C" void kernel_launch(void* const* d_in, const int* in_sizes, int n_in,
                              void* d_out, int out_size, void* d_ws, size_t ws_size,
                              hipStream_t stream) {
    const float*     x  = (const float*)d_in[0];
    const long long* ei = (const long long*)d_in[1];   // int64 edge_index [2,E]
    const float*     ew = (const float*)d_in[2];
    const float*     W1 = (const float*)d_in[3];
    const float*     b1 = (const float*)d_in[4];
    const float*     W2 = (const float*)d_in[5];
    const float*     b2 = (const float*)d_in[6];
    float*           out = (float*)d_out;

    const int N = N_NODES, E = N_EDGES;
    const int ND = N * DIM;                    // 6,400,000

    float* deg = (float*)d_ws;                 // N   (becomes dinv)
    float* sn  = deg + N;                      // N   self-loop norm
    float* nrm = sn + N;                       // E   edge norm
    float* hg  = nrm + E;                      // N*D
    float* acc = hg + (size_t)ND;              // N*D

    const int T = 256;
    dim3 blk(T);

    // --- gcn_norm ---
    k_deg_init <<<(N + T - 1) / T, blk, 0, stream>>>(deg, N);
    k_deg_edges<<<(E + T - 1) / T, blk, 0, stream>>>(ei, ew, deg, E);
    k_dinv     <<<(N + T - 1) / T, blk, 0, stream>>>(deg, sn, N);
    k_norm     <<<(E + T - 1) / T, blk, 0, stream>>>(ei, ew, deg, nrm, E);

    // --- layer 1 ---
    gemm16_wmma<<<N / 16, blk, 0, stream>>>(x, W1, hg);            // hg = x@W1
    k_zero     <<<(ND + T - 1) / T, blk, 0, stream>>>(acc, ND);
    k_aggregate<<<((E * 32) + T - 1) / T, blk, 0, stream>>>(hg, ei, nrm, acc, E);
    k_bias_relu<<<(ND + T - 1) / T, blk, 0, stream>>>(hg, acc, sn, b1, ND);

    // --- layer 2 + dropout ---
    gemm16_wmma<<<N / 16, blk, 0, stream>>>(hg, W2, acc);          // acc = h1@W2
    k_zero     <<<(ND + T - 1) / T, blk, 0, stream>>>(hg, ND);
    k_aggregate<<<((E * 32) + T - 1) / T, blk, 0, stream>>>(acc, ei, nrm, hg, E);
    k_final    <<<(ND + T - 1) / T, blk, 0, stream>>>(out, hg, acc, sn, b2, ND);
}